// Trans4map_knowledge_17832704213138
// MI455X (gfx1250) — compile-verified
//
#include <hip/hip_runtime.h>
#include <hip/hip_bf16.h>

typedef __attribute__((ext_vector_type(2))) float v2f;
typedef __attribute__((ext_vector_type(8))) float v8f;

#define MAP_CELLS 250000            // 500*500
#define EGO_H 1024
#define EGO_W 2048
#define HW (EGO_H * EGO_W)          // 2097152
#define FEAT_H 128
#define FEAT_W 256
#define FEAT_C 64
#define MEM_DIM 256
#define PIX_PER_THREAD 8
#define PIX_PER_BLOCK 2048          // 256 threads * 8
#define NSCANBLK (HW / PIX_PER_BLOCK) // 1024

// ---------------- workspace layout (int32 view) ----------------
// ws[0]               : max(proj)  (threshold)
// ws[1]               : total inlier count
// ws[2 .. 2+1024)     : per-block inlier counts -> exclusive offsets (in place)
// ws[1026 .. 1026+HW) : positions[] (stable argsort of ~mask)
// total ~ 8.4 MB

__global__ void k_init(int* ws) {
    if (threadIdx.x == 0) ws[0] = (int)0x80000000;
}

__global__ void k_max(const int* __restrict__ proj, int n, int* __restrict__ ws) {
    __shared__ int sm[256];
    int t = threadIdx.x;
    int m = (int)0x80000000;
    for (int i = blockIdx.x * blockDim.x + t; i < n; i += gridDim.x * blockDim.x)
        m = max(m, proj[i]);
    sm[t] = m; __syncthreads();
    for (int s = 128; s > 0; s >>= 1) {
        if (t < s) sm[t] = max(sm[t], sm[t + s]);
        __syncthreads();
    }
    if (t == 0) atomicMax(&ws[0], sm[0]);
}

__global__ void k_count(const int* __restrict__ mask, int* __restrict__ blockCounts) {
    __shared__ int sm[256];
    int t = threadIdx.x;
    int base = blockIdx.x * PIX_PER_BLOCK + t * PIX_PER_THREAD;
    int c = 0;
    #pragma unroll
    for (int i = 0; i < PIX_PER_THREAD; i++) c += (mask[base + i] != 0);
    sm[t] = c; __syncthreads();
    for (int s = 128; s > 0; s >>= 1) {
        if (t < s) sm[t] += sm[t + s];
        __syncthreads();
    }
    if (t == 0) blockCounts[blockIdx.x] = sm[0];
}

// single block of 1024 threads: in-place exclusive scan of 1024 counts
__global__ void k_scan(int* __restrict__ blockOff, int* __restrict__ total) {
    __shared__ int sm[1024];
    int t = threadIdx.x;
    int v = blockOff[t];
    sm[t] = v; __syncthreads();
    for (int off = 1; off < 1024; off <<= 1) {
        int add = (t >= off) ? sm[t - off] : 0;
        __syncthreads();
        sm[t] += add;
        __syncthreads();
    }
    blockOff[t] = sm[t] - v;                // exclusive
    if (t == 1023) *total = sm[t];          // grand total inliers
}

__global__ void k_scatter(const int* __restrict__ mask,
                          const int* __restrict__ blockOff,
                          const int* __restrict__ totalp,
                          int* __restrict__ positions) {
    __shared__ int sm[256];
    int t = threadIdx.x;
    int base = blockIdx.x * PIX_PER_BLOCK + t * PIX_PER_THREAD;
    int c = 0;
    #pragma unroll
    for (int i = 0; i < PIX_PER_THREAD; i++) c += (mask[base + i] != 0);
    sm[t] = c; __syncthreads();
    int v = c;
    for (int off = 1; off < 256; off <<= 1) {
        int add = (t >= off) ? sm[t - off] : 0;
        __syncthreads();
        sm[t] += add;
        __syncthreads();
    }
    int total = *totalp;
    int run = blockOff[blockIdx.x] + (sm[t] - v);  // inliers strictly before pixel `base`
    #pragma unroll
    for (int i = 0; i < PIX_PER_THREAD; i++) {
        int p = base + i;
        if (mask[p] != 0) { positions[run] = p; run++; }
        else             { positions[total + (p - run)] = p; }
    }
}

// Main: gather (on-the-fly bilinear) -> WMMA f32 16x16x4 linear layer -> masked store.
// One wave = 16 map cells. A = W_lin tile (M = out-channel), B = gathered^T (N = cell).
// Output stores are non-temporal: 256 MB write-once stream must not evict the
// L2-resident working set (features 8 MB, positions 8 MB, W_lin 64 KB).
__global__ __launch_bounds__(256) void k_main(
        const float* __restrict__ feat,      // [64,128,256]
        const int*   __restrict__ proj,      // [250000]
        const float* __restrict__ Wl,        // [256,64] row-major
        const float* __restrict__ bl,        // [256]
        const int*   __restrict__ ws_scalars,// ws[0] = threshold
        const int*   __restrict__ positions, // [HW]
        float*       __restrict__ out)       // [256*250000] memory, then [250000] mask
{
    __shared__ float ldsA[8][16 * FEAT_C];   // per-wave 16 cells x 64 ch
    __shared__ float ldsM[8][16];            // per-wave observed flags

    const int t = threadIdx.x;
    const int wave = t >> 5;
    const int lane = t & 31;
    const int cellbase = (blockIdx.x * 8 + wave) * 16;
    if (cellbase >= MAP_CELLS) return;       // uniform per wave -> EXEC stays all-1s

    const int threshold = ws_scalars[0];

    // ---- gather phase: 2 lanes per cell, 32 channels each ----
    const int cell = lane >> 1;
    const int half = lane & 1;
    int q = proj[cellbase + cell];
    int mflag = (q < threshold) ? 1 : 0;
    int qc = min(max(q, 0), HW - 1);
    int p = positions[qc];
    int py = p >> 11;            // /2048
    int px = p & 2047;
    float ys = (float)py * ((float)(FEAT_H - 1) / (float)(EGO_H - 1));
    float xs = (float)px * ((float)(FEAT_W - 1) / (float)(EGO_W - 1));
    int y0 = (int)ys; int y1 = min(y0 + 1, FEAT_H - 1); float wy = ys - (float)y0;
    int x0 = (int)xs; int x1 = min(x0 + 1, FEAT_W - 1); float wx = xs - (float)x0;
    float w00 = (1.f - wy) * (1.f - wx), w01 = (1.f - wy) * wx;
    float w10 = wy * (1.f - wx),         w11 = wy * wx;
    const float* f00 = feat + y0 * FEAT_W + x0;
    const float* f01 = feat + y0 * FEAT_W + x1;
    const float* f10 = feat + y1 * FEAT_W + x0;
    const float* f11 = feat + y1 * FEAT_W + x1;
    #pragma unroll 4
    for (int c = 0; c < 32; c++) {
        int ch = half * 32 + c;
        long o = (long)ch * (FEAT_H * FEAT_W);
        ldsA[wave][cell * FEAT_C + ch] =
            w00 * f00[o] + w01 * f01[o] + w10 * f10[o] + w11 * f11[o];
    }
    if (half == 0) {
        float mf = mflag ? 1.0f : 0.0f;
        ldsM[wave][cell] = mf;
        __builtin_nontemporal_store(mf, &out[(long)MEM_DIM * MAP_CELLS + cellbase + cell]);
    }
    __syncthreads();   // safe: terminated waves count as signaled

    // ---- matmul phase ----
    const float* As = ldsA[wave];
    const int nloc = lane & 15;              // N = cell-in-tile, also M-row select for A
    const int ksel = (lane < 16) ? 0 : 2;    // A/B half-wave K offset per ISA layout
    const int mof  = (lane < 16) ? 0 : 8;    // C/D half-wave M offset
    const float mf = ldsM[wave][nloc];
    const long col = (long)cellbase + nloc;

    for (int j = 0; j < 16; j++) {           // 16 out-channel tiles of 16
        const int mrow0 = j * 16 + mof;
        v8f acc;
        #pragma unroll
        for (int v = 0; v < 8; v++) acc[v] = bl[mrow0 + v];   // C seeded with bias
        const float* wrowBase = Wl + (j * 16 + nloc) * FEAT_C;
        #pragma unroll
        for (int ks = 0; ks < 16; ks++) {
            int k0 = ks * 4 + ksel;
            v2f a, b;
            a[0] = wrowBase[k0];
            a[1] = wrowBase[k0 + 1];
            b[0] = As[nloc * FEAT_C + k0];
            b[1] = As[nloc * FEAT_C + k0 + 1];
            acc = __builtin_amdgcn_wmma_f32_16x16x4_f32(
                      false, a, false, b, (short)0, acc, false, false);
        }
        #pragma unroll
        for (int v = 0; v < 8; v++)
            __builtin_nontemporal_store(acc[v] * mf,
                                        &out[(long)(mrow0 + v) * MAP_CELLS + col]);
    }
}

extern "C" void kernel_launch(void* const* d_in, const int* in_sizes, int n_in,
                              void* d_out, int out_size, void* d_ws, size_t ws_size,
                              hipStream_t stream) {
    const float* features = (const float*)d_in[0];   // [1,64,128,256] f32
    const int*   proj     = (const int*)d_in[1];     // [250000] (int)
    const int*   mask     = (const int*)d_in[2];     // [1024,2048] bool->int
    const float* Wl       = (const float*)d_in[3];   // [256,64] f32
    const float* bl       = (const float*)d_in[4];   // [256] f32
    float* out = (float*)d_out;

    int* wsI       = (int*)d_ws;
    int* totalSlot = wsI + 1;
    int* blockOff  = wsI + 2;
    int* positions = wsI + 1026;

    k_init   <<<1, 1, 0, stream>>>(wsI);
    k_max    <<<256, 256, 0, stream>>>(proj, MAP_CELLS, wsI);
    k_count  <<<NSCANBLK, 256, 0, stream>>>(mask, blockOff);
    k_scan   <<<1, 1024, 0, stream>>>(blockOff, totalSlot);
    k_scatter<<<NSCANBLK, 256, 0, stream>>>(mask, blockOff, totalSlot, positions);

    const int nWaves  = MAP_CELLS / 16;          // 15625
    const int nBlocks = (nWaves + 7) / 8;        // 1954
    k_main   <<<nBlocks, 256, 0, stream>>>(features, proj, Wl, bl, wsI, positions, out);
}